// GCNConv_27848567947398
// MI455X (gfx1250) — compile-verified
//
#include <hip/hip_runtime.h>

typedef __attribute__((ext_vector_type(2))) float v2f;
typedef __attribute__((ext_vector_type(8))) float v8f;

#define N_NODES   100000
#define N_EDGES   1600000
#define FEATS     128
#define EDGE_DIM  7
#define TILES_PER_BLOCK 8   // edge tiles (16 edges each) per block

// ---------------------------------------------------------------------------
// Kernel 1: x = feat @ fc_w^T  via V_WMMA_F32_16X16X4_F32
// Block = 256 threads = 8 waves; block owns a 16-row strip of feat,
// wave w owns output columns [16w, 16w+16). K-loop: 128/4 = 32 WMMAs/wave.
// ---------------------------------------------------------------------------
__global__ __launch_bounds__(256) void k_gemm_x(const float* __restrict__ feat,
                                                const float* __restrict__ fc_w,
                                                float* __restrict__ x) {
  __shared__ float At[16 * FEATS];  // 8 KB staging of the A tile
  const int row0 = blockIdx.x * 16;
  const float4* g4 = (const float4*)(feat + (size_t)row0 * FEATS);
  float4* l4 = (float4*)At;
  for (int i = threadIdx.x; i < 16 * FEATS / 4; i += 256) l4[i] = g4[i];
  __syncthreads();

  const int lane = threadIdx.x & 31;
  const int wave = threadIdx.x >> 5;
  const int m  = lane & 15;    // A row / B column within tile
  const int hi = lane >> 4;    // K phase: 0 -> K=k,k+1 ; 1 -> K=k+2,k+3
  const int nb = wave * 16;
  const int nc = nb + m;       // global output column this lane serves for B

  v8f c = {};
  for (int k = 0; k < FEATS; k += 4) {
    const int ka = k + hi * 2;
    v2f a, b;
    a.x = At[m * FEATS + ka];          // A[m][ka], A[m][ka+1]
    a.y = At[m * FEATS + ka + 1];
    b.x = fc_w[nc * FEATS + ka];       // B[ka][nc] = fc_w[nc][ka]
    b.y = fc_w[nc * FEATS + ka + 1];
    c = __builtin_amdgcn_wmma_f32_16x16x4_f32(false, a, false, b,
                                              (short)0, c, false, false);
  }
  // D layout: VGPR r holds row M = r + 8*hi, col N = lane%16
  float* o = x + (size_t)row0 * FEATS + nb;
#pragma unroll
  for (int r = 0; r < 8; ++r) o[(r + 8 * hi) * FEATS + m] = c[r];
}

// ---------------------------------------------------------------------------
// Degree pipeline
// ---------------------------------------------------------------------------
__global__ void k_deg_init(float* __restrict__ deg) {
  int i = blockIdx.x * 256 + threadIdx.x;
  if (i < N_NODES) deg[i] = 1.0f;              // the "+ 1.0"
}

__global__ void k_deg_count(const int* __restrict__ dst, float* __restrict__ deg) {
  int i = blockIdx.x * 256 + threadIdx.x;
  if (i < N_EDGES) atomicAdd(&deg[dst[i]], 1.0f);
}

__global__ void k_dinv(const float* __restrict__ deg, float* __restrict__ dinv) {
  int i = blockIdx.x * 256 + threadIdx.x;
  if (i < N_NODES) dinv[i] = 1.0f / sqrtf(deg[i]);
}

// ---------------------------------------------------------------------------
// out = relu(x + root_emb) / deg   (edge kernel then accumulates on top)
// ---------------------------------------------------------------------------
__global__ void k_out_init(const float* __restrict__ x,
                           const float* __restrict__ root_emb,
                           const float* __restrict__ deg,
                           float* __restrict__ out) {
  size_t i = (size_t)blockIdx.x * 256 + threadIdx.x;
  if (i < (size_t)N_NODES * FEATS) {
    int n = (int)(i >> 7);        // /128
    int cch = (int)(i & 127);
    out[i] = fmaxf(x[i] + root_emb[cch], 0.0f) / deg[n];
  }
}

// ---------------------------------------------------------------------------
// Kernel 2 (the heavy one): fused edge linear + gather + relu + scatter-add.
// Per wave: B-frags of edge_w^T (K padded 7->8, zero fill) hoisted out of the
// loop; per 16-edge tile: A-frag from edge_feat, C initialized to edge_b,
// 2x V_WMMA_F32_16X16X4_F32, then coalesced gather of x[src], relu, norm
// scale and global_atomic_add_f32 into out[dst].
// ---------------------------------------------------------------------------
__global__ __launch_bounds__(256) void k_edge(const float* __restrict__ edge_feat,
                                              const int* __restrict__ src,
                                              const int* __restrict__ dst,
                                              const float* __restrict__ edge_w,
                                              const float* __restrict__ edge_b,
                                              const float* __restrict__ x,
                                              const float* __restrict__ dinv,
                                              float* __restrict__ out) {
  const int lane = threadIdx.x & 31;
  const int wave = threadIdx.x >> 5;
  const int m  = lane & 15;
  const int hi = lane >> 4;
  const int nb = wave * 16;
  const int nc = nb + m;          // output channel this lane serves in B/D

  // B-frags: B[k][n] = edge_w[nc][k]; pad k==7 with zero.
  v2f b0, b1;
  {
    const int j0 = hi * 2;        // 0 or 2  (always < 7)
    b0.x = edge_w[nc * EDGE_DIM + j0];
    b0.y = edge_w[nc * EDGE_DIM + j0 + 1];
    const int j1 = 4 + hi * 2;    // 4 or 6
    b1.x = edge_w[nc * EDGE_DIM + j1];
    b1.y = (j1 + 1 < EDGE_DIM) ? edge_w[nc * EDGE_DIM + j1 + 1] : 0.0f;
  }
  const float bias = edge_b[nc];

  const long e0base = (long)blockIdx.x * (16 * TILES_PER_BLOCK);
  for (int t = 0; t < TILES_PER_BLOCK; ++t) {
    const long e0 = e0base + t * 16;
    const float* efrow = edge_feat + (e0 + m) * EDGE_DIM;
    v2f a0, a1;
    {
      const int j0 = hi * 2;
      a0.x = efrow[j0];
      a0.y = efrow[j0 + 1];
      const int j1 = 4 + hi * 2;
      a1.x = efrow[j1];
      a1.y = (j1 + 1 < EDGE_DIM) ? efrow[j1 + 1] : 0.0f;
    }
    v8f c = {bias, bias, bias, bias, bias, bias, bias, bias};
    c = __builtin_amdgcn_wmma_f32_16x16x4_f32(false, a0, false, b0,
                                              (short)0, c, false, false);
    c = __builtin_amdgcn_wmma_f32_16x16x4_f32(false, a1, false, b1,
                                              (short)0, c, false, false);

    // D row r+8*hi = edge e0+r+8*hi, col nc. Lanes 0..15 cover 16 contiguous
    // channels of one edge row -> 64B coalesced gather + 64B coalesced atomics.
#pragma unroll
    for (int r = 0; r < 8; ++r) {
      const long e = e0 + r + 8 * hi;
      const int s = src[e];
      const int d = dst[e];
      const float nrm = dinv[s] * dinv[d];
      float val = c[r] + x[(size_t)s * FEATS + nc];
      val = fmaxf(val, 0.0f) * nrm;
      atomicAdd(out + (size_t)d * FEATS + nc, val);
    }
  }
}

// ---------------------------------------------------------------------------
extern "C" void kernel_launch(void* const* d_in, const int* in_sizes, int n_in,
                              void* d_out, int out_size, void* d_ws, size_t ws_size,
                              hipStream_t stream) {
  const float* feat      = (const float*)d_in[0];
  const float* edge_feat = (const float*)d_in[1];
  const int*   src       = (const int*)d_in[2];
  const int*   dst       = (const int*)d_in[3];
  const float* fc_w      = (const float*)d_in[4];
  const float* edge_w    = (const float*)d_in[5];
  const float* edge_b    = (const float*)d_in[6];
  const float* root_emb  = (const float*)d_in[7];
  float* out = (float*)d_out;

  float* x    = (float*)d_ws;                    // N_NODES * 128 floats
  float* deg  = x + (size_t)N_NODES * FEATS;     // N_NODES floats
  float* dinv = deg + N_NODES;                   // N_NODES floats

  // 1) x = feat @ fc_w^T   (WMMA f32, 6250 blocks of 8 waves)
  k_gemm_x<<<N_NODES / 16, 256, 0, stream>>>(feat, fc_w, x);

  // 2) degree + deg^-0.5
  k_deg_init <<<(N_NODES + 255) / 256, 256, 0, stream>>>(deg);
  k_deg_count<<<(N_EDGES + 255) / 256, 256, 0, stream>>>(dst, deg);
  k_dinv     <<<(N_NODES + 255) / 256, 256, 0, stream>>>(deg, dinv);

  // 3) out = relu(x + root) / deg  (self term; edges accumulate on top)
  k_out_init<<<(int)(((size_t)N_NODES * FEATS + 255) / 256), 256, 0, stream>>>(
      x, root_emb, deg, out);

  // 4) fused edge linear + message passing (WMMA f32 + f32 atomics)
  k_edge<<<N_EDGES / (16 * TILES_PER_BLOCK), 256, 0, stream>>>(
      edge_feat, src, dst, edge_w, edge_b, x, dinv, out);
}